// VGAEncoder_26061861552454
// MI455X (gfx1250) — compile-verified
//
#include <hip/hip_runtime.h>

#define D 128
#define LAT 64
#define EPSBN 1e-5f
#define SLOPE 0.01f
#define KCHUNK 32
#define LDSTR 36   // KCHUNK + 4 pad dwords -> conflict-free b64 reads (36*i mod 64 distinct)

typedef __attribute__((ext_vector_type(2))) float v2f;
typedef __attribute__((ext_vector_type(4))) float v4f;
typedef __attribute__((ext_vector_type(8))) float v8f;

// ---------------------------------------------------------------------------
// WMMA GEMM: H[n x ocols] = X[n x 128] @ W[128 x ocols]   (f32, 16x16x4 WMMA)
// Block = 256 threads (8 waves). Each wave computes a 16x32 output tile
// (2 accumulators sharing one A fragment). W is staged transposed into LDS
// in 32-deep K chunks; each B fragment is then one ds_load_b64.
//   A layout (32-bit 16x4): lanes 0-15 rows M=0..15 hold K,K+1; lanes 16-31 hold K+2,K+3
//   B layout mirrors C/D striping (cols across lanes, K pair across vgpr/half-wave)
//   C/D: vgpr v, lanes 0-15 -> (M=v, N=lane); lanes 16-31 -> (M=v+8, N=lane-16)
// ---------------------------------------------------------------------------
__global__ __launch_bounds__(256) void gemm_wmma_f32(
    const float* __restrict__ X, const float* __restrict__ W,
    float* __restrict__ H, int nrows, int ocols, int oshift, int cgShift)
{
    __shared__ float WT[D * LDSTR];               // max 128 cols * 36 dwords = 18 KB

    const int wave = threadIdx.x >> 5;
    const int lane = threadIdx.x & 31;
    const int colGroups = 1 << cgShift;           // 4 (ocols=128) or 2 (ocols=64)
    const int cg = wave & (colGroups - 1);
    const int rg = wave >> cgShift;               // 0..(8/colGroups - 1)
    const int rowsPerBlock = (8 >> cgShift) * 16; // 32 or 64

    const int row0 = blockIdx.x * rowsPerBlock + rg * 16;
    const int col0 = cg * 32;
    const bool active = (row0 + 16 <= nrows);     // wave-uniform

    const int l15 = lane & 15;
    const int hi  = lane >> 4;

    const float* __restrict__ arow = X + (size_t)(row0 + l15) * D;
    const int stageN = KCHUNK << oshift;          // 32 * ocols

    v8f acc0 = {};
    v8f acc1 = {};

    for (int kc = 0; kc < D; kc += KCHUNK) {
        __syncthreads();                          // protect WT readers of prev chunk
        // stage W[kc..kc+31][*] transposed: WT[c*36 + kk] = W[kc+kk][c]
        for (int i = threadIdx.x; i < stageN; i += 256) {
            const int kk = i >> oshift;
            const int c  = i & (ocols - 1);
            WT[c * LDSTR + kk] = W[(size_t)(kc + kk) * ocols + c];
        }
        __syncthreads();

        if (active) {
            const float* __restrict__ a0 = arow + kc;
            #pragma unroll
            for (int k = 0; k < KCHUNK; k += 4) {
                const int ka = k + 2 * hi;
                v2f a  = *(const v2f*)(a0 + ka);                           // global b64
                v2f b0 = *(const v2f*)(&WT[(col0 +      l15) * LDSTR + ka]); // ds b64
                v2f b1 = *(const v2f*)(&WT[(col0 + 16 + l15) * LDSTR + ka]); // ds b64
                acc0 = __builtin_amdgcn_wmma_f32_16x16x4_f32(
                    false, a, false, b0, (short)0, acc0, false, false);
                acc1 = __builtin_amdgcn_wmma_f32_16x16x4_f32(
                    false, a, false, b1, (short)0, acc1, false, false);
            }
        }
    }

    if (active) {
        float* __restrict__ out0 = H + (size_t)(row0 + hi * 8) * ocols + col0 + l15;
        #pragma unroll
        for (int v = 0; v < 8; ++v) {
            out0[(size_t)v * ocols]      = acc0[v];
            out0[(size_t)v * ocols + 16] = acc1[v];
        }
    }
}

// ---------------------------------------------------------------------------
// Degree / normalization kernels
// ---------------------------------------------------------------------------
__global__ void deg_init(float* __restrict__ deg, int n) {
    int i = blockIdx.x * blockDim.x + threadIdx.x;
    if (i < n) deg[i] = 1.0f;                     // self-loop contribution
}

__global__ void deg_edges(const int* __restrict__ dst, float* __restrict__ deg, int e) {
    int i = blockIdx.x * blockDim.x + threadIdx.x;
    if (i < e) atomicAdd(&deg[dst[i]], 1.0f);
}

__global__ void deg_to_dinv(float* __restrict__ deg, int n) {
    int i = blockIdx.x * blockDim.x + threadIdx.x;
    if (i < n) deg[i] = rsqrtf(fmaxf(deg[i], 1e-12f));
}

// out[i,c] = h[i,c]*dinv[i]^2 + bias[c]   (self-loop message + post-agg bias)
__global__ void self_init(const float* __restrict__ h, const float* __restrict__ dinv,
                          const float* __restrict__ bias, float* __restrict__ out,
                          int total, int dshift, int dmask) {
    int idx = blockIdx.x * blockDim.x + threadIdx.x;
    if (idx >= total) return;
    int i = idx >> dshift;
    int c = idx & dmask;
    float w = dinv[i];
    out[idx] = h[idx] * w * w + bias[c];
}

// out[dst,c] += h[src,c] * dinv[src]*dinv[dst], 4 channels per thread
__global__ void edge_scatter(const float* __restrict__ h, const float* __restrict__ dinv,
                             const int* __restrict__ srcs, const int* __restrict__ dsts,
                             float* __restrict__ out, long long total,
                             int qshift, int qmask, int d) {
    long long idx = (long long)blockIdx.x * blockDim.x + threadIdx.x;
    if (idx >= total) return;
    int e  = (int)(idx >> qshift);
    int c4 = ((int)idx & qmask) << 2;
    int s = srcs[e];
    int t = dsts[e];
    float nrm = dinv[s] * dinv[t];
    v4f hv = *(const v4f*)(h + (size_t)s * d + c4);
    float* o = out + (size_t)t * d + c4;
    atomicAdd(o + 0, hv.x * nrm);
    atomicAdd(o + 1, hv.y * nrm);
    atomicAdd(o + 2, hv.z * nrm);
    atomicAdd(o + 3, hv.w * nrm);
}

// ---------------------------------------------------------------------------
// BatchNorm (training-mode batch stats) + leaky-relu + residual
// ---------------------------------------------------------------------------
__global__ void bn_zero(float* __restrict__ stats) { stats[threadIdx.x] = 0.0f; }

__global__ void bn_stats(const float* __restrict__ x, float* __restrict__ stats, int n) {
    int c = threadIdx.x;                           // 128 threads = one channel each
    float s = 0.0f, s2 = 0.0f;
    for (int r = blockIdx.x; r < n; r += gridDim.x) {
        float v = x[(size_t)r * D + c];
        s += v;
        s2 += v * v;
    }
    atomicAdd(&stats[c], s);
    atomicAdd(&stats[D + c], s2);
}

__global__ void bn_apply(const float* __restrict__ agg, const float* __restrict__ stats,
                         const float* __restrict__ gamma, const float* __restrict__ beta,
                         float* __restrict__ x, int total, float inv_n) {
    int idx = blockIdx.x * blockDim.x + threadIdx.x;
    if (idx >= total) return;
    int c = idx & (D - 1);
    float mean = stats[c] * inv_n;
    float var  = stats[D + c] * inv_n - mean * mean;   // biased var, like jnp.var
    float y = (agg[idx] - mean) * rsqrtf(var + EPSBN) * gamma[c] + beta[c];
    y = (y >= 0.0f) ? y : SLOPE * y;
    x[idx] = y + x[idx];                               // skip connection (in place)
}

// ---------------------------------------------------------------------------
extern "C" void kernel_launch(void* const* d_in, const int* in_sizes, int n_in,
                              void* d_out, int out_size, void* d_ws, size_t ws_size,
                              hipStream_t stream) {
    const float* x     = (const float*)d_in[0];
    const int*   ei    = (const int*)  d_in[1];
    const float* Ws    = (const float*)d_in[2];
    const float* bs    = (const float*)d_in[3];
    const float* Wmu   = (const float*)d_in[4];
    const float* bmu   = (const float*)d_in[5];
    const float* Wlv   = (const float*)d_in[6];
    const float* blv   = (const float*)d_in[7];
    const float* gamma = (const float*)d_in[8];
    const float* beta  = (const float*)d_in[9];

    const int n = in_sizes[0] / D;
    const int E = in_sizes[1] / 2;
    const int* src = ei;
    const int* dst = ei + E;

    const size_t nd = (size_t)n * D;
    float* bufX  = (float*)d_ws;           // current features  [n x 128]
    float* bufH  = bufX + nd;              // gemm output       [n x 128]
    float* bufA  = bufH + nd;              // aggregation out   [n x 128]
    float* dinv  = bufA + nd;              // [n]
    float* stats = dinv + n;               // [256]

    const int tpb = 256;
    const int ndTotal = (int)nd;

    // --- degree -> dinv -------------------------------------------------
    deg_init   <<<(n + tpb - 1) / tpb, tpb, 0, stream>>>(dinv, n);
    deg_edges  <<<(E + tpb - 1) / tpb, tpb, 0, stream>>>(dst, dinv, E);
    deg_to_dinv<<<(n + tpb - 1) / tpb, tpb, 0, stream>>>(dinv, n);

    // --- x -> workspace (inputs must stay unmodified) -------------------
    hipMemcpyAsync(bufX, x, nd * sizeof(float), hipMemcpyDeviceToDevice, stream);

    const long long eTotal128 = (long long)E * (D / 4);

    // --- 3 GCN layers ----------------------------------------------------
    for (int layer = 0; layer < 3; ++layer) {
        gemm_wmma_f32<<<(n + 31) / 32, tpb, 0, stream>>>(
            bufX, Ws + (size_t)layer * D * D, bufH, n, D, /*oshift=*/7, /*cgShift=*/2);

        self_init<<<(ndTotal + tpb - 1) / tpb, tpb, 0, stream>>>(
            bufH, dinv, bs + (size_t)layer * D, bufA, ndTotal, 7, D - 1);

        edge_scatter<<<(int)((eTotal128 + tpb - 1) / tpb), tpb, 0, stream>>>(
            bufH, dinv, src, dst, bufA, eTotal128, 5, 31, D);

        bn_zero <<<1, 256, 0, stream>>>(stats);
        bn_stats<<<256, D, 0, stream>>>(bufA, stats, n);
        bn_apply<<<(ndTotal + tpb - 1) / tpb, tpb, 0, stream>>>(
            bufA, stats, gamma, beta, bufX, ndTotal, 1.0f / (float)n);
    }

    // --- heads: mu and logvar (L = 64) -----------------------------------
    float* muOut = (float*)d_out;
    float* lvOut = (float*)d_out + (size_t)n * LAT;
    const int nlTotal = n * LAT;
    const long long eTotal64 = (long long)E * (LAT / 4);

    // mu
    gemm_wmma_f32<<<(n + 63) / 64, tpb, 0, stream>>>(bufX, Wmu, bufH, n, LAT, 6, 1);
    self_init<<<(nlTotal + tpb - 1) / tpb, tpb, 0, stream>>>(bufH, dinv, bmu, muOut, nlTotal, 6, LAT - 1);
    edge_scatter<<<(int)((eTotal64 + tpb - 1) / tpb), tpb, 0, stream>>>(
        bufH, dinv, src, dst, muOut, eTotal64, 4, 15, LAT);

    // logvar
    gemm_wmma_f32<<<(n + 63) / 64, tpb, 0, stream>>>(bufX, Wlv, bufH, n, LAT, 6, 1);
    self_init<<<(nlTotal + tpb - 1) / tpb, tpb, 0, stream>>>(bufH, dinv, blv, lvOut, nlTotal, 6, LAT - 1);
    edge_scatter<<<(int)((eTotal64 + tpb - 1) / tpb), tpb, 0, stream>>>(
        bufH, dinv, src, dst, lvOut, eTotal64, 4, 15, LAT);
}